// DynamicConvolutionAttention_81449759801718
// MI455X (gfx1250) — compile-verified
//
#include <hip/hip_runtime.h>
#include <math.h>

// Problem constants (match reference)
#define B_     256
#define T_     2048
#define QDIM   1024
#define ADIM   128
#define DC     8
#define DK     21
#define SC     8
#define SK     21
#define PL     11
#define PAD    10          // (DK-1)/2 == PL-1
#define NG     (DC * DK)   // 168
#define TCHUNK 512

typedef float v2f __attribute__((ext_vector_type(2)));
typedef float v8f __attribute__((ext_vector_type(8)));

#if __has_builtin(__builtin_amdgcn_global_load_async_to_lds_b32) && \
    __has_builtin(__builtin_amdgcn_s_wait_asynccnt)
#define HAVE_ASYNC_LDS 1
typedef __attribute__((address_space(1))) int* glob_i32_ptr;
typedef __attribute__((address_space(3))) int* lds_i32_ptr;
#endif

__device__ __forceinline__ float fast_tanh(float x) {
#if __has_builtin(__builtin_amdgcn_tanhf)
  return __builtin_amdgcn_tanhf(x);       // hardware V_TANH_F32 on gfx1250
#else
  return tanhf(x);
#endif
}

__device__ __forceinline__ v8f vzero8() {
  v8f z = {0.f, 0.f, 0.f, 0.f, 0.f, 0.f, 0.f, 0.f};
  return z;
}

// Full-precision f32 WMMA: D(16x16) = A(16x4) * B(4x16) + C
// A frag (v2f): lane L -> row M = L&15, K = (L>>4)*2 + {0,1}
// B frag (v2f): lane L -> col N = L&15, K = (L>>4)*2 + {0,1}
// C/D (v8f):    lanes 0-15 rows 0-7, lanes 16-31 rows 8-15, col = L&15
__device__ __forceinline__ v8f wmma4(v2f a, v2f b, v8f c) {
  return __builtin_amdgcn_wmma_f32_16x16x4_f32(false, a, false, b, (short)0, c,
                                               false, false);
}

// ---------------------------------------------------------------------------
// Kernel 1: H = tanh(query @ W_w.T + W_b)   (256 x 128, K=1024)
// 128 tiles of 16x16; 1 wave per tile; 8 waves per block -> 16 blocks.
// ---------------------------------------------------------------------------
__global__ void k_hidden(const float* __restrict__ q, const float* __restrict__ Ww,
                         const float* __restrict__ Wb, float* __restrict__ H) {
  const int lane = threadIdx.x & 31;
  const int wave = threadIdx.x >> 5;
  const int tile = blockIdx.x * 8 + wave;      // 0..127
  const int m0 = (tile >> 3) * 16;             // 16 m-tiles
  const int n0 = (tile & 7) * 16;              // 8 n-tiles
  const int li = lane & 15;
  const int kh = (lane >> 4) * 2;

  const float* qrow = q + (m0 + li) * QDIM + kh;
  const float* wrow = Ww + (n0 + li) * QDIM + kh;   // W_w.T[k][n] = W_w[n][k]

  v8f acc = vzero8();
  for (int k0 = 0; k0 < QDIM; k0 += 4) {
    v2f a;  a.x = qrow[k0];  a.y = qrow[k0 + 1];
    v2f bb; bb.x = wrow[k0]; bb.y = wrow[k0 + 1];
    acc = wmma4(a, bb, acc);
  }

  const int ncol = n0 + li;
  const float bias = Wb[ncol];
  const int rbase = m0 + (lane >> 4) * 8;
#pragma unroll
  for (int i = 0; i < 8; ++i)
    H[(rbase + i) * ADIM + ncol] = fast_tanh(acc[i] + bias);
}

// ---------------------------------------------------------------------------
// Kernel 2: G = H @ V_w.T   (256 x 168, K=128). 16 x 11 tiles (N padded to 176).
// ---------------------------------------------------------------------------
__global__ void k_gdyn(const float* __restrict__ H, const float* __restrict__ Vw,
                       float* __restrict__ G) {
  const int lane = threadIdx.x & 31;
  const int wave = threadIdx.x >> 5;
  const int tile = blockIdx.x * 8 + wave;      // 0..175
  const int m0 = (tile / 11) * 16;
  const int n0 = (tile % 11) * 16;
  const int li = lane & 15;
  const int kh = (lane >> 4) * 2;

  const int ncol = n0 + li;
  const bool valid = (ncol < NG);
  const float vm = valid ? 1.f : 0.f;          // mask via select, keep EXEC full

  const float* hrow = H + (m0 + li) * ADIM + kh;
  const float* vrow = Vw + (valid ? ncol : 0) * ADIM + kh;

  v8f acc = vzero8();
  for (int k0 = 0; k0 < ADIM; k0 += 4) {
    v2f a;  a.x = hrow[k0];       a.y = hrow[k0 + 1];
    v2f bb; bb.x = vrow[k0] * vm; bb.y = vrow[k0 + 1] * vm;
    acc = wmma4(a, bb, acc);
  }

  const int rbase = m0 + (lane >> 4) * 8;
  if (valid) {
#pragma unroll
    for (int i = 0; i < 8; ++i)
      G[(rbase + i) * NG + ncol] = acc[i];
  }
}

// ---------------------------------------------------------------------------
// Kernel 3 (fused core): per (b, t) compute f (static conv), g (dynamic conv),
// h = tanh([f|g] @ [U|T].T + T_b), e = v . h   -> E[b,t]
// One block per (b, 512-t chunk); 8 waves, each wave owns 16-t tiles.
// The (16t x 16feat) A-matrix feeds 8 N-tiles x 4 K-steps of f32 WMMA.
// LDS layouts are paired so hot-loop fragment reads are single ds_load_b64.
// ---------------------------------------------------------------------------
__global__ void k_energy(const float* __restrict__ align_, const float* __restrict__ Gd,
                         const float* __restrict__ Fw, const float* __restrict__ Uw,
                         const float* __restrict__ Tw, const float* __restrict__ Tb,
                         const float* __restrict__ vw, float* __restrict__ E) {
  __shared__ __align__(16) float sx[TCHUNK + 2 * PAD];   // padded alignment segment
  // sWp[p][n][j] = Wcat[2p+j][n], Wcat[k][n] = k<8 ? U_w[n][k] : T_w[n][k-8]
  __shared__ __align__(16) float sWp[8 * ADIM * 2];
  // sFGp[k][c] = c<8 ? F_w[c][k] : G[b][c-8][k]
  __shared__ __align__(16) float sFGp[DK * 16];
  __shared__ float sv[ADIM];
  __shared__ float sTb[ADIM];

  const int b = blockIdx.y;
  const int t0 = blockIdx.x * TCHUNK;
  const int tid = threadIdx.x;

  // ---- stage alignment chunk (async copy to LDS on gfx1250) ----
  for (int i = tid; i < TCHUNK + 2 * PAD; i += 256) {
    int gi = t0 + i - PAD;
#ifdef HAVE_ASYNC_LDS
    if (gi >= 0 && gi < T_) {
      __builtin_amdgcn_global_load_async_to_lds_b32(
          (glob_i32_ptr)(align_ + b * T_ + gi),
          (lds_i32_ptr)&sx[i], 0, 0);
    } else {
      sx[i] = 0.f;
    }
#else
    sx[i] = (gi >= 0 && gi < T_) ? align_[b * T_ + gi] : 0.f;
#endif
  }
  // ---- stage weights ----
  for (int i = tid; i < 8 * ADIM * 2; i += 256) {
    int p = i >> 8, rem = i & 255, n = rem >> 1, j = rem & 1;
    int k = 2 * p + j;
    sWp[i] = (k < SC) ? Uw[n * SC + k] : Tw[n * DC + (k - SC)];
  }
  for (int i = tid; i < DK * 16; i += 256) {
    int k = i >> 4, c = i & 15;
    sFGp[i] = (c < SC) ? Fw[c * SK + k] : Gd[b * NG + (c - SC) * DK + k];
  }
  for (int i = tid; i < ADIM; i += 256) { sv[i] = vw[i]; sTb[i] = Tb[i]; }
#ifdef HAVE_ASYNC_LDS
  __builtin_amdgcn_s_wait_asynccnt(0);
#endif
  __syncthreads();

  const int lane = tid & 31;
  const int wave = tid >> 5;
  const int li = lane & 15;
  const int kh = (lane >> 4) * 2;   // K offset within a 4-wide K step
  const int khp = lane >> 4;        // which K-pair of the step

  for (int it = 0; it < TCHUNK / (16 * 8); ++it) {
    const int lt0 = (it * 8 + wave) * 16;   // local t base of this 16-row tile
    const int lm = lt0 + li;                // this lane's local t (A row)

    // Build A fragments: A[t][c], c = 0..7 static f, 8..15 dynamic g
    v2f af[4];
#pragma unroll
    for (int jf = 0; jf < 4; ++jf) {
      const int c0 = jf * 4 + kh;           // even channel index
      float a0 = 0.f, a1 = 0.f;
#pragma unroll
      for (int k = 0; k < DK; ++k) {
        const float x = sx[lm + k];         // windows[b, t, k] = x_pad[t+k]
        const v2f w = *(const v2f*)&sFGp[k * 16 + c0];   // one ds_load_b64
        a0 = fmaf(x, w.x, a0);
        a1 = fmaf(x, w.y, a1);
      }
      af[jf].x = a0; af[jf].y = a1;
    }

    float part[8];
#pragma unroll
    for (int i = 0; i < 8; ++i) part[i] = 0.f;

    for (int nt = 0; nt < ADIM / 16; ++nt) {
      const int ncol = nt * 16 + li;
      v8f acc = vzero8();
#pragma unroll
      for (int jf = 0; jf < 4; ++jf) {
        const v2f bb = *(const v2f*)&sWp[(jf * 2 + khp) * (ADIM * 2) + ncol * 2];
        acc = wmma4(af[jf], bb, acc);
      }
      const float bias = sTb[ncol];
      const float vn = sv[ncol];
#pragma unroll
      for (int i = 0; i < 8; ++i)
        part[i] = fmaf(vn, fast_tanh(acc[i] + bias), part[i]);
    }

    // Reduce over the 16 lanes holding the same row group (offsets < 16 stay in-half)
#pragma unroll
    for (int off = 8; off >= 1; off >>= 1) {
#pragma unroll
      for (int i = 0; i < 8; ++i) part[i] += __shfl_xor(part[i], off, 32);
    }

    float* erow = E + b * T_ + t0 + lt0;
    if (lane == 0) {
#pragma unroll
      for (int i = 0; i < 8; ++i) erow[i] = part[i];      // rows 0..7
    }
    if (lane == 16) {
#pragma unroll
      for (int i = 0; i < 8; ++i) erow[8 + i] = part[i];  // rows 8..15
    }
  }
}

// ---------------------------------------------------------------------------
// Kernel 4: prior (beta-binomial taps) + log + stable row softmax over T.
// One block per batch row; each thread holds 8 energies in registers.
// ---------------------------------------------------------------------------
__global__ void k_softmax(const float* __restrict__ E, const float* __restrict__ align_,
                          const float* __restrict__ P, float* __restrict__ out) {
  __shared__ float sP[PL];
  __shared__ float sred[8];
  const int b = blockIdx.x;
  const int tid = threadIdx.x;

  __builtin_prefetch(align_ + b * T_ + tid * 8, 0, 1);   // global_prefetch on row

  if (tid < PL) sP[tid] = P[tid];
  __syncthreads();

  float et[T_ / 256];
  float m = -3.0e38f;
#pragma unroll
  for (int j = 0; j < T_ / 256; ++j) {
    const int t = tid + j * 256;
    float p = 0.f;
#pragma unroll
    for (int k = 0; k < PL; ++k) {
      const int gi = t + k - PAD;                        // x_pad[t+k]
      const float x = (gi >= 0 && gi < T_) ? align_[b * T_ + gi] : 0.f;
      p = fmaf(x, sP[k], p);
    }
    const float e = E[b * T_ + t] + logf(fmaxf(p, 1e-6f));
    et[j] = e;
    m = fmaxf(m, e);
  }

#pragma unroll
  for (int off = 16; off >= 1; off >>= 1) m = fmaxf(m, __shfl_xor(m, off, 32));
  if ((tid & 31) == 0) sred[tid >> 5] = m;
  __syncthreads();
  float M = sred[0];
#pragma unroll
  for (int i = 1; i < 8; ++i) M = fmaxf(M, sred[i]);
  __syncthreads();

  float s = 0.f;
#pragma unroll
  for (int j = 0; j < T_ / 256; ++j) s += expf(et[j] - M);
#pragma unroll
  for (int off = 16; off >= 1; off >>= 1) s += __shfl_xor(s, off, 32);
  if ((tid & 31) == 0) sred[tid >> 5] = s;
  __syncthreads();
  float S = 0.f;
#pragma unroll
  for (int i = 0; i < 8; ++i) S += sred[i];
  const float inv = 1.0f / S;

#pragma unroll
  for (int j = 0; j < T_ / 256; ++j)
    out[b * T_ + tid + j * 256] = expf(et[j] - M) * inv;
}

// ---------------------------------------------------------------------------
extern "C" void kernel_launch(void* const* d_in, const int* in_sizes, int n_in,
                              void* d_out, int out_size, void* d_ws, size_t ws_size,
                              hipStream_t stream) {
  const float* query = (const float*)d_in[0];
  const float* align_ = (const float*)d_in[1];
  const float* P     = (const float*)d_in[2];
  const float* Ww    = (const float*)d_in[3];
  const float* Wb    = (const float*)d_in[4];
  const float* Vw    = (const float*)d_in[5];
  const float* Fw    = (const float*)d_in[6];   // (SC,1,SK)
  const float* Uw    = (const float*)d_in[7];
  const float* Tw    = (const float*)d_in[8];
  const float* Tb    = (const float*)d_in[9];
  const float* vw    = (const float*)d_in[10];  // (1, ADIM)
  float* out = (float*)d_out;

  // Workspace: H (256x128) | G (256x168) | E (256x2048)  ~= 2.3 MB
  float* H  = (float*)d_ws;
  float* Gd = H + B_ * ADIM;
  float* E  = Gd + B_ * NG;

  k_hidden<<<dim3((B_ / 16) * (ADIM / 16) / 8), 256, 0, stream>>>(query, Ww, Wb, H);
  k_gdyn<<<dim3(16 * 11 / 8), 256, 0, stream>>>(H, Vw, Gd);
  k_energy<<<dim3(T_ / TCHUNK, B_), 256, 0, stream>>>(align_, Gd, Fw, Uw, Tw, Tb, vw, E);
  k_softmax<<<dim3(B_), 256, 0, stream>>>(E, align_, P, out);
}